// BatchShapingLoss_1666447311265
// MI455X (gfx1250) — compile-verified
//
#include <hip/hip_runtime.h>
#include <cmath>

#define NPTS  1000
#define NROWS 512
#define NCOLS 128
#define EPSF  1e-10f

typedef __attribute__((ext_vector_type(2))) float v2f;
typedef __attribute__((ext_vector_type(8))) float v8f;

// pdf(g) = exp2( (a-1)*log2(g) + (b-1)*log2(1-g) + c2 ),  c2 = -BETALN/ln2
__device__ __forceinline__ float beta_pdf(float g, float c2) {
  float l0 = __builtin_amdgcn_logf(g);          // v_log_f32  (log2)
  float l1 = __builtin_amdgcn_logf(1.0f - g);   // v_log_f32
  float arg = fmaf(-0.4f, l0, fmaf(-0.6f, l1, c2));
  return __builtin_amdgcn_exp2f(arg);           // v_exp_f32
}

// ---------------- Kernel 1: bitonic sort of each column (512 elems) --------
__global__ void sort_cols(const float* __restrict__ x, float* __restrict__ sorted) {
  __shared__ float lds[NROWS];
  const int c = blockIdx.x;
  const int t = threadIdx.x;                 // 256 threads, 2 elems each
  lds[t]       = x[t * NCOLS + c];
  lds[t + 256] = x[(t + 256) * NCOLS + c];
  for (int k = 2; k <= NROWS; k <<= 1) {
    for (int j = k >> 1; j > 0; j >>= 1) {
      __syncthreads();
      int i = ((t & ~(j - 1)) << 1) | (t & (j - 1));  // bit j of i is 0
      int p = i | j;
      bool up = ((i & k) == 0);
      float a = lds[i];
      float b = lds[p];
      if ((a > b) == up) { lds[i] = b; lds[p] = a; }
    }
  }
  __syncthreads();
  // column-major output: sorted[c*512 + r]
  sorted[c * NROWS + t]       = lds[t];
  sorted[c * NROWS + t + 256] = lds[t + 256];
}

// ------------- Kernel 2: trapezoid integral, 4 chunks per element ----------
// block = 256 threads = 64 elements x 4 chunks of the 999-step grid
__global__ void integrate(const float* __restrict__ sorted,
                          float* __restrict__ sqerr, float c2) {
  __shared__ float part[256];
  const int tid = threadIdx.x;
  const int el  = tid & 63;
  const int q   = tid >> 6;
  const int e   = blockIdx.x * 64 + el;
  const float sx    = sorted[e];
  const float scale = (sx - EPSF) * (1.0f / (float)(NPTS - 1)); // uniform dg
  int i0 = 1 + q * 250;
  int i1 = i0 + 250; if (i1 > NPTS) i1 = NPTS;   // i = 1..999 overall
  float fi  = (float)i0;
  float sum = 0.0f;
#pragma unroll 4
  for (int i = i0; i < i1; ++i) {
    float g = fmaf(fi, scale, EPSF);
    sum += beta_pdf(g, c2);
    fi += 1.0f;
  }
  part[tid] = sum;
  __syncthreads();
  if (tid < 64) {
    float tot = part[tid] + part[tid + 64] + part[tid + 128] + part[tid + 192];
    float p1  = beta_pdf(EPSF + scale, c2);   // first grid point (i=1)
    float pL  = beta_pdf(sx, c2);             // last grid point  (i=999)
    float pcdf = scale * (tot - 0.5f * (p1 + pL));
    int   r    = e & (NROWS - 1);             // row index within column
    float ecdf = (float)(r + 1) * (1.0f / (float)(NROWS + 1));
    float d = pcdf - ecdf;
    sqerr[e] = d * d;
  }
}

// --------- Kernel 3: WMMA-based total reduction of 65536 floats ------------
// B = ones => D[m,n] = sum_k A[m,k] + C[m,n]; f32 accumulate, full precision.
__global__ void reduce_wmma(const float* __restrict__ in,
                            float* __restrict__ out, float inv_n) {
  const int tid  = threadIdx.x;     // 256 threads = 8 waves
  const int wave = tid >> 5;
  const int lane = tid & 31;
  v8f c = {0.f, 0.f, 0.f, 0.f, 0.f, 0.f, 0.f, 0.f};
  v2f ones; ones.x = 1.0f; ones.y = 1.0f;
  const float* base = in + wave * 8192;       // 65536 / 8 waves
  for (int it = 0; it < 128; ++it) {          // 64 values per WMMA
    v2f a = *(const v2f*)(base + it * 64 + lane * 2);
    c = __builtin_amdgcn_wmma_f32_16x16x4_f32(
        /*neg_a=*/false, a, /*neg_b=*/false, ones,
        /*c_mod=*/(short)0, c, /*reuse_a=*/false, /*reuse_b=*/false);
  }
  // every row-sum is replicated across 16 columns of D: sum all, divide by 16
  float s = c[0] + c[1] + c[2] + c[3] + c[4] + c[5] + c[6] + c[7];
  for (int off = 16; off > 0; off >>= 1)
    s += __shfl_xor(s, off, 32);
  __shared__ float wsum[8];
  if (lane == 0) wsum[wave] = s * 0.0625f;
  __syncthreads();
  if (tid == 0) {
    float t = 0.0f;
    for (int w = 0; w < 8; ++w) t += wsum[w];
    out[0] = t * inv_n;
  }
}

extern "C" void kernel_launch(void* const* d_in, const int* in_sizes, int n_in,
                              void* d_out, int out_size, void* d_ws, size_t ws_size,
                              hipStream_t stream) {
  (void)in_sizes; (void)n_in; (void)out_size; (void)ws_size;
  const float* x = (const float*)d_in[0];
  float* sorted = (float*)d_ws;                 // 65536 floats
  float* sqerr  = sorted + NROWS * NCOLS;       // 65536 floats
  float* out    = (float*)d_out;

  const double betaln = lgamma(0.6) + lgamma(0.4) - lgamma(1.0);
  const float  c2     = (float)(-betaln / 0.6931471805599453); // -BETALN/ln2

  sort_cols<<<NCOLS, 256, 0, stream>>>(x, sorted);
  integrate<<<(NROWS * NCOLS) / 64, 256, 0, stream>>>(sorted, sqerr, c2);
  reduce_wmma<<<1, 256, 0, stream>>>(sqerr, out, 1.0f / (float)NROWS);
}